// GAEncode_3332894621720
// MI455X (gfx1250) — compile-verified
//
#include <hip/hip_runtime.h>
#include <hip/hip_bf16.h>

typedef __attribute__((ext_vector_type(2))) float v2f;
typedef __attribute__((ext_vector_type(8))) float v8f;

#define BB   16
#define NN   4096
#define CC   64
#define KK   32
#define CHUNK 512                 // columns per LDS chunk
#define NCHUNK (NN / CHUNK)       // 8
#define NEG_INF (-3.4e38f)

// ---------------------------------------------------------------------------
// Kernel 1: KNN top-32 per row via V_WMMA_F32_16X16X4_F32 score tiles.
// Score s[i,j] = 2*xi.xj - ||xj||^2  (row-constant -||xi||^2 dropped; does not
// change per-row top-k ranking).  A row = [2x,2y,2z,1], B col = [x,y,z,-|p|^2].
// Column chunk (x,y,z planes, 6KB) is staged into LDS with gfx1250 async
// global->LDS copies (ASYNCcnt), fragments built branch-free from LDS.
// Grid: B * (N/16) workgroups, 256 threads (8 waves) each.
// ---------------------------------------------------------------------------
__global__ __launch_bounds__(256) void knn_wmma_kernel(const float* __restrict__ xyz,
                                                       int* __restrict__ idx_out) {
    const int b  = blockIdx.x / (NN / 16);
    const int rt = blockIdx.x % (NN / 16);
    const float* X = xyz + (size_t)b * 3 * NN;   // x:[0..N), y:[N..2N), z:[2N..3N)

    __shared__ float sm  [16 * CHUNK];           // 32 KB scores / candidate vals
    __shared__ int   smi [16 * CHUNK];           // 32 KB candidate indices
    __shared__ float cols[3 * CHUNK];            // 6 KB staged x|y|z planes

    const int tid  = threadIdx.x;
    const int lane = tid & 31;
    const int wave = tid >> 5;
    const int half = lane >> 4;                  // 0: holds K=0,1   1: holds K=2,3
    const int lm   = lane & 15;

    // --- A fragment (constant across all column tiles), branch-free --------
    const int row = rt * 16 + lm;
    const float ax = X[row];
    const float ay = X[NN + row];
    const float az = X[2 * NN + row];
    v2f afrag;
    afrag.x = half ? 2.0f * az : 2.0f * ax;
    afrag.y = half ? 1.0f      : 2.0f * ay;

    // private sorted (descending) top-32 of this thread's column stripe
    float vals[KK];
    int   inds[KK];
    for (int i = 0; i < KK; ++i) { vals[i] = NEG_INF; inds[i] = 0; }

    const int r = tid >> 4;      // row (0..15) this thread scans
    const int t = tid & 15;      // 32-column stripe within the chunk

    for (int chunk = 0; chunk < NCHUNK; ++chunk) {
        // ---- async-stage this chunk's x/y/z columns into LDS --------------
        // 3 planes * 512 dwords, 256 threads -> 6 async B32 copies per thread
        for (int a = 0; a < 3; ++a) {
            for (int i = tid; i < CHUNK; i += 256) {
                const float* gsrc = X + (size_t)a * NN + chunk * CHUNK + i;
                const unsigned ldst = (unsigned)(size_t)&cols[a * CHUNK + i];
                asm volatile("global_load_async_to_lds_b32 %0, %1, off"
                             :: "v"(ldst), "v"(gsrc) : "memory");
            }
        }
        asm volatile("s_wait_asynccnt 0x0" ::: "memory");
        __syncthreads();

        // ---- fill 16 x 512 score tile: 8 waves * 4 WMMAs ------------------
        for (int j = 0; j < 4; ++j) {
            const int ctl = wave * 4 + j;                 // local col tile 0..31
            const int cl  = ctl * 16 + lm;                // column within chunk
            const float cx = cols[cl];
            const float cy = cols[CHUNK + cl];
            const float cz = cols[2 * CHUNK + cl];
            const float q  = -(cx * cx + cy * cy + cz * cz);
            v2f bfrag;                                    // branch-free select
            bfrag.x = half ? cz : cx;
            bfrag.y = half ? q  : cy;

            v8f acc = {};
            acc = __builtin_amdgcn_wmma_f32_16x16x4_f32(
                /*neg_a=*/false, afrag, /*neg_b=*/false, bfrag,
                /*c_mod=*/(short)0, acc, /*reuse_a=*/false, /*reuse_b=*/false);
            // D layout: VGPR g -> M = g + 8*half, N = lm
#pragma unroll
            for (int g = 0; g < 8; ++g) {
                const int M = g + 8 * half;
                sm[M * CHUNK + ctl * 16 + lm] = acc[g];
            }
        }
        __syncthreads();

        // ---- scan my 32-column stripe, maintain sorted top-32 -------------
        for (int i = 0; i < 32; ++i) {
            const float v = sm[r * CHUNK + t * 32 + i];
            if (v > vals[KK - 1]) {
                int p = KK - 1;
                while (p > 0 && vals[p - 1] < v) {
                    vals[p] = vals[p - 1];
                    inds[p] = inds[p - 1];
                    --p;
                }
                vals[p] = v;
                inds[p] = chunk * CHUNK + t * 32 + i;
            }
        }
        __syncthreads();   // before next chunk overwrites cols/sm
    }

    // ---- publish each thread's sorted list, then 16-way merge per row -----
    for (int s = 0; s < KK; ++s) {
        sm [r * CHUNK + t * 32 + s] = vals[s];
        smi[r * CHUNK + t * 32 + s] = inds[s];
    }
    __syncthreads();

    if (t == 0) {
        int heads[16];
        for (int q = 0; q < 16; ++q) heads[q] = 0;
        int* out = idx_out + ((size_t)b * NN + (size_t)rt * 16 + r) * KK;
        for (int kk = 0; kk < KK; ++kk) {
            float best = NEG_INF;
            int   bq   = 0;
            for (int q = 0; q < 16; ++q) {
                if (heads[q] < KK) {
                    const float v = sm[r * CHUNK + q * 32 + heads[q]];
                    if (v > best) { best = v; bq = q; }
                }
            }
            out[kk] = smi[r * CHUNK + bq * 32 + heads[bq]];
            heads[bq]++;
        }
    }
}

// ---------------------------------------------------------------------------
// Kernel 0: zero the sigma accumulator (workspace), deterministic each launch.
// ---------------------------------------------------------------------------
__global__ void zero_sigma_kernel(float* __restrict__ sig) {
    if (threadIdx.x == 0 && blockIdx.x == 0) sig[0] = 0.0f;
}

// ---------------------------------------------------------------------------
// Kernel 2: global sum of offset^2 (for sigma). One block per (b,n).
// ---------------------------------------------------------------------------
__global__ __launch_bounds__(256) void sigma_kernel(const float* __restrict__ feats,
                                                    const int* __restrict__ idx,
                                                    float* __restrict__ sig_accum) {
    const int bn = blockIdx.x;
    const int b  = bn / NN;
    const int n  = bn % NN;
    const float* selfrow = feats + ((size_t)b * NN + n) * CC;
    const int*   id      = idx   + ((size_t)b * NN + n) * KK;

    float s = 0.0f;
    for (int e = threadIdx.x; e < KK * CC; e += 256) {
        const int k = e >> 6;      // /CC
        const int c = e & (CC - 1);
        const float o = feats[((size_t)b * NN + id[k]) * CC + c] - selfrow[c];
        s += o * o;
    }

    __shared__ float red[256];
    red[threadIdx.x] = s;
    __syncthreads();
    for (int st = 128; st > 0; st >>= 1) {
        if (threadIdx.x < st) red[threadIdx.x] += red[threadIdx.x + st];
        __syncthreads();
    }
    if (threadIdx.x == 0) atomicAdd(sig_accum, red[0]);
}

// ---------------------------------------------------------------------------
// Kernel 3: gaed = offset/(sigma+eps)*alpha + beta ; max-pool over k.
// 64 threads per (b,n) (one per channel), 4 points per 256-thread block.
// ---------------------------------------------------------------------------
__global__ __launch_bounds__(256) void pool_kernel(const float* __restrict__ feats,
                                                   const int* __restrict__ idx,
                                                   const float* __restrict__ alpha,
                                                   const float* __restrict__ beta,
                                                   const float* __restrict__ sig_accum,
                                                   float* __restrict__ out) {
    const int g = blockIdx.x * 4 + (threadIdx.x >> 6);   // flat (b*N+n)
    const int c = threadIdx.x & (CC - 1);
    const int b = g / NN;
    const int n = g % NN;

    const float sigma = sig_accum[0] * (1.0f / ((float)BB * NN * KK * CC));
    const float inv   = 1.0f / (sigma + 1e-5f);

    const float selfv = feats[((size_t)b * NN + n) * CC + c];
    const float al    = alpha[c] * inv;
    const float be    = beta[c];
    const int*  id    = idx + ((size_t)b * NN + n) * KK;

    float m = NEG_INF;
    for (int k = 0; k < KK; ++k) {
        if (k + 1 < KK) {
            // gfx1250 global_prefetch_b8 of next neighbor row
            __builtin_prefetch(&feats[((size_t)b * NN + id[k + 1]) * CC + c], 0, 0);
        }
        const float o = feats[((size_t)b * NN + id[k]) * CC + c] - selfv;
        const float v = o * al + be;
        m = fmaxf(m, v);
    }
    out[((size_t)b * NN + n) * CC + c] = m;
}

// ---------------------------------------------------------------------------
extern "C" void kernel_launch(void* const* d_in, const int* in_sizes, int n_in,
                              void* d_out, int out_size, void* d_ws, size_t ws_size,
                              hipStream_t stream) {
    const float* xyz   = (const float*)d_in[0];   // [B,3,N]
    const float* feats = (const float*)d_in[1];   // [B,N,C]
    const float* alpha = (const float*)d_in[2];   // [C]
    const float* beta  = (const float*)d_in[3];   // [C]
    float* out = (float*)d_out;                   // [B,N,C]

    // workspace layout: [0, B*N*K*4) neighbor indices, then sigma accumulator
    int*   ws_idx   = (int*)d_ws;
    float* ws_sigma = (float*)((char*)d_ws + (size_t)BB * NN * KK * sizeof(int));

    zero_sigma_kernel<<<1, 64, 0, stream>>>(ws_sigma);

    // KNN: B * N/16 row tiles
    knn_wmma_kernel<<<BB * (NN / 16), 256, 0, stream>>>(xyz, ws_idx);

    // sigma partial sums: one block per (b,n)
    sigma_kernel<<<BB * NN, 256, 0, stream>>>(feats, ws_idx, ws_sigma);

    // normalize + affine + max-pool: 4 points per block
    pool_kernel<<<(BB * NN) / 4, 256, 0, stream>>>(feats, ws_idx, alpha, beta,
                                                   ws_sigma, out);
}